// Hausdorff_39737037423050
// MI455X (gfx1250) — compile-verified
//
#include <hip/hip_runtime.h>
#include <hip/hip_bf16.h>

typedef __attribute__((ext_vector_type(2))) float v2f;
typedef __attribute__((ext_vector_type(8))) float v8f;

#define HG 96
#define HW (HG * HG)          // 9216
#define NTILES (HW / 16)      // 576 column/row tiles
#define NBATCH 8
#define BIGF 1.0e9f

// ---------------------------------------------------------------------------
// Init: zero the 8 per-sample max accumulators (float 0.0 == 0x0 bits).
// ---------------------------------------------------------------------------
__global__ void haus_init(unsigned int* wsMax) {
    int t = threadIdx.x;
    if (t < NBATCH) wsMax[t] = 0u;
}

// ---------------------------------------------------------------------------
// Main: one wave per (sample, 16-row tile). WMMA f32 16x16x4 computes a
// 16x16 block of squared pairwise grid distances per instruction.
// ---------------------------------------------------------------------------
__global__ void haus_main(const float* __restrict__ pred,
                          const float* __restrict__ targ,
                          unsigned int* __restrict__ wsMax) {
    const int lane  = threadIdx.x & 31;
    const int wid   = threadIdx.x >> 5;                 // 4 waves / block
    const int task  = blockIdx.x * 4 + wid;             // 0 .. 8*576-1
    const int n     = task / NTILES;                    // sample
    const int Itile = task - n * NTILES;                // row tile
    const int Ibase = Itile * 16;

    const int laneN = lane & 15;
    const int half  = lane >> 4;                        // 0: lanes 0-15, 1: 16-31
    const float inv = 1.0f / (float)HG;

    const float* pr = pred + n * HW;
    const float* tg = targ + n * HW;

    // ---- A fragment (16x4, K split across lane halves), built once --------
    {
    }
    const int p  = Ibase + laneN;
    const float xp = (float)(p / HG) * inv;
    const float yp = (float)(p % HG) * inv;
    const float sp = xp * xp + yp * yp;
    v2f aF;
    aF.x = half ? (-2.0f * xp) : sp;
    aF.y = half ? (-2.0f * yp) : 1.0f;

    // ---- running masked minima (C/D layout: elem v, half h -> row v+8h) ---
    v8f accA, accB;                                     // distA: tgt=b ; distB: tgt=a
#pragma unroll
    for (int v = 0; v < 8; ++v) { accA[v] = 3.0e38f; accB[v] = 3.0e38f; }

    const v8f czero = {};

    for (int J = 0; J < NTILES; ++J) {
        const int q  = J * 16 + laneN;
        const float xq = (float)(q / HG) * inv;
        const float yq = (float)(q % HG) * inv;
        const float sq = xq * xq + yq * yq;

        // B fragment (4x16): VGPR v, half h -> K = v + 2h
        v2f bF;
        bF.x = half ? xq : 1.0f;
        bF.y = half ? yq : sq;

        // d2[v] = squared distance, rows Ibase+v+8h, col q
        v8f d2 = __builtin_amdgcn_wmma_f32_16x16x4_f32(
            /*neg_a=*/false, aF, /*neg_b=*/false, bF,
            /*c_mod=*/(short)0, czero, /*reuse_a=*/false, /*reuse_b=*/false);

        // target masks for this column: bias excluded columns by +BIG
        const bool am = pr[q] > 0.5f;                   // point in A
        const bool bm = tg[q] > 0.5f;                   // point in B
        const float biasA = bm ? 0.0f : BIGF;           // distA targets set B
        const float biasB = am ? 0.0f : BIGF;           // distB targets set A

#pragma unroll
        for (int v = 0; v < 8; ++v) {
            accA[v] = fminf(accA[v], d2[v] + biasA);
            accB[v] = fminf(accB[v], d2[v] + biasB);
        }
    }

    // ---- min over columns: butterfly across the 16 lanes of each half -----
#pragma unroll
    for (int off = 1; off < 16; off <<= 1) {
#pragma unroll
        for (int v = 0; v < 8; ++v) {
            accA[v] = fminf(accA[v], __shfl_xor(accA[v], off, 32));
            accB[v] = fminf(accB[v], __shfl_xor(accB[v], off, 32));
        }
    }

    // ---- apply source masks, take max over the 8 rows this half owns ------
    float lmax = 0.0f;                                  // empty source -> 0
#pragma unroll
    for (int v = 0; v < 8; ++v) {
        const int rowg = Ibase + v + 8 * half;
        const bool ar = pr[rowg] > 0.5f;
        const bool br = tg[rowg] > 0.5f;
        if (ar && !br) lmax = fmaxf(lmax, accA[v]);     // src A\B, tgt B
        if (br && !ar) lmax = fmaxf(lmax, accB[v]);     // src B\A, tgt A
    }

    // lmax >= 0, so float bits are uint-monotone
    if (laneN == 0) atomicMax(&wsMax[n], __float_as_uint(lmax));
}

// ---------------------------------------------------------------------------
// Final: mean over batch of sqrt(max squared Hausdorff).
// ---------------------------------------------------------------------------
__global__ void haus_final(const unsigned int* __restrict__ wsMax,
                           float* __restrict__ out) {
    if (threadIdx.x == 0) {
        float s = 0.0f;
        for (int n = 0; n < NBATCH; ++n)
            s += sqrtf(__uint_as_float(wsMax[n]));
        out[0] = s * (1.0f / (float)NBATCH);
    }
}

extern "C" void kernel_launch(void* const* d_in, const int* in_sizes, int n_in,
                              void* d_out, int out_size, void* d_ws, size_t ws_size,
                              hipStream_t stream) {
    const float* pred = (const float*)d_in[0];   // predict [8,1,96,96] f32
    const float* targ = (const float*)d_in[1];   // target  [8,1,96,96] f32
    float* out = (float*)d_out;                  // scalar f32
    unsigned int* wsMax = (unsigned int*)d_ws;   // 8 uints of scratch

    haus_init<<<1, 32, 0, stream>>>(wsMax);

    const int waves  = NBATCH * NTILES;          // 4608 wave tasks
    const int blocks = waves / 4;                // 128 threads = 4 waves/block
    haus_main<<<blocks, 128, 0, stream>>>(pred, targ, wsMax);

    haus_final<<<1, 32, 0, stream>>>(wsMax, out);
}